// ReadHead_34557306864267
// MI455X (gfx1250) — compile-verified
//
#include <hip/hip_runtime.h>
#include <math.h>

#define EPSF 1e-8f
#define B_ 128
#define N_ 512
#define W_ 64

typedef __attribute__((ext_vector_type(2))) float v2f;
typedef __attribute__((ext_vector_type(8))) float v8f;

// ---------------------------------------------------------------------------
// Kernel 1: content addressing.
//   sim[b][n] = dot(memory[b][n]/(||memory[b][n]||+eps), key/(|key|+eps))
//   cw[b][:]  = softmax(sim * (1 + softplus(r_beta[b])))
// One block per batch, 512 threads (one per memory row).
// ---------------------------------------------------------------------------
__global__ __launch_bounds__(512) void content_kernel(
    const float* __restrict__ r_key, const float* __restrict__ r_beta,
    const float* __restrict__ memory, float* __restrict__ cw)
{
    const int b = blockIdx.x;
    const int n = threadIdx.x;
    __shared__ float ks[W_];
    __shared__ float red[N_];

    if (n < W_) {
        float kv = r_key[b * W_ + n];
        ks[n] = kv / (fabsf(kv) + EPSF);
    }
    __syncthreads();

    const float4* mr = (const float4*)(memory + ((size_t)b * N_ + n) * W_);
    float ss = 0.f, dt = 0.f;
#pragma unroll
    for (int i = 0; i < W_ / 4; ++i) {
        float4 v = mr[i];
        ss += v.x * v.x + v.y * v.y + v.z * v.z + v.w * v.w;
        dt += v.x * ks[4 * i + 0] + v.y * ks[4 * i + 1] +
              v.z * ks[4 * i + 2] + v.w * ks[4 * i + 3];
    }
    // beta = 1 + softplus(r_beta), numerically stable
    float x = r_beta[b];
    float beta = 1.f + fmaxf(x, 0.f) + log1pf(expf(-fabsf(x)));
    float s = (dt / (sqrtf(ss) + EPSF)) * beta;

    // block softmax over N
    red[n] = s;
    __syncthreads();
    for (int off = N_ / 2; off > 0; off >>= 1) {
        if (n < off) red[n] = fmaxf(red[n], red[n + off]);
        __syncthreads();
    }
    float mx = red[0];
    __syncthreads();
    float e = expf(s - mx);
    red[n] = e;
    __syncthreads();
    for (int off = N_ / 2; off > 0; off >>= 1) {
        if (n < off) red[n] += red[n + off];
        __syncthreads();
    }
    cw[(size_t)b * N_ + n] = e / red[0];
}

// ---------------------------------------------------------------------------
// Kernel 2: batched dual mat-vec via V_WMMA_F32_16X16X4_F32 (full f32).
//   dir=0: f[i] = sum_j L[i][j] * w[j]   (A = L tile, B = w replicated)
//   dir=1: b[i] = sum_j L[j][i] * w[j]   (A = w replicated, B = L tile)
// One wave32 per 16-wide output tile; 128 K-steps of 4.
// A 16x4 f32 layout: lanes 0-15 row M, v0=K(2*half+0), v1=K(2*half+1).
// B 4x16 f32 layout: lanes % 16 = col N, v0=K(2*half+0), v1=K(2*half+1).
// ---------------------------------------------------------------------------
__global__ __launch_bounds__(256) void link_matvec_kernel(
    const float* __restrict__ link, const float* __restrict__ rw,
    float* __restrict__ fw, float* __restrict__ bw)
{
    const int bid  = blockIdx.x;
    const int b    = bid >> 3;
    const int rem  = bid & 7;
    const int dir  = rem >> 2;        // 0 = forward, 1 = backward
    const int tg   = rem & 3;
    const int wave = threadIdx.x >> 5;
    const int lane = threadIdx.x & 31;
    const int half = lane >> 4;       // 0: lanes 0-15, 1: lanes 16-31
    const int lm   = lane & 15;
    const int i0   = (tg * 8 + wave) * 16;

    const float* L = link + (size_t)b * N_ * N_;
    const float* w = rw + (size_t)b * N_;
    v8f acc = {};

    if (dir == 0) {
        // forward: A tile = L[i0+lm][j + 2*half .. +1], B = w broadcast
        const float* arow = L + (size_t)(i0 + lm) * N_ + 2 * half;
        const float* wrep = w + 2 * half;
#pragma unroll 4
        for (int j = 0; j < N_; j += 4) {
            v2f a  = *(const v2f*)(arow + j);
            v2f bm = *(const v2f*)(wrep + j);
            acc = __builtin_amdgcn_wmma_f32_16x16x4_f32(
                false, a, false, bm, (short)0, acc, false, false);
        }
        // every column n of C holds f[i0+m]; C row M = r + 8*half at lane lm
        if (lm == 0) {
            float* out = fw + (size_t)b * N_ + i0 + 8 * half;
#pragma unroll
            for (int r = 0; r < 8; ++r) out[r] = acc[r];
        }
    } else {
        // backward: B tile = L[j + 2*half + {0,1}][i0+lm], A = w broadcast
        const float* bcol = L + (size_t)(2 * half) * N_ + i0 + lm;
        const float* wrep = w + 2 * half;
#pragma unroll 4
        for (int j = 0; j < N_; j += 4) {
            v2f a  = *(const v2f*)(wrep + j);
            v2f bm;
            bm.x = bcol[(size_t)j * N_];
            bm.y = bcol[(size_t)(j + 1) * N_];
            acc = __builtin_amdgcn_wmma_f32_16x16x4_f32(
                false, a, false, bm, (short)0, acc, false, false);
        }
        // every row of C holds b[i0+n]; take acc[0], col N = lm
        if (half == 0) bw[(size_t)b * N_ + i0 + lm] = acc[0];
    }
}

// ---------------------------------------------------------------------------
// Kernel 3: mode blend + weighted read-out.
//   weights = m0*b + m1*c + m2*f ; read[b][w] = sum_n weights[n]*memory[b][n][w]
// One block per batch, 256 threads.
// ---------------------------------------------------------------------------
__global__ __launch_bounds__(256) void combine_kernel(
    const float* __restrict__ r_mode, const float* __restrict__ memory,
    const float* __restrict__ cw, const float* __restrict__ fw,
    const float* __restrict__ bw, float* __restrict__ out_read,
    float* __restrict__ out_weights)
{
    const int b = blockIdx.x;
    const int t = threadIdx.x;
    __shared__ float wsh[N_];
    __shared__ float pr[256];

    float m0r = r_mode[b * 3 + 0];
    float m1r = r_mode[b * 3 + 1];
    float m2r = r_mode[b * 3 + 2];
    float mmax = fmaxf(m0r, fmaxf(m1r, m2r));
    float e0 = expf(m0r - mmax), e1 = expf(m1r - mmax), e2 = expf(m2r - mmax);
    float inv = 1.f / (e0 + e1 + e2);
    float m0 = e0 * inv, m1 = e1 * inv, m2 = e2 * inv;

    const size_t base = (size_t)b * N_;
    for (int n = t; n < N_; n += 256) {
        float wv = m0 * bw[base + n] + m1 * cw[base + n] + m2 * fw[base + n];
        wsh[n] = wv;
        out_weights[base + n] = wv;
    }
    __syncthreads();

    const int w = t & 63;
    const int g = t >> 6;
    const float* mb = memory + (size_t)b * N_ * W_;
    float acc = 0.f;
    for (int n = g * 128; n < g * 128 + 128; ++n)
        acc += wsh[n] * mb[(size_t)n * W_ + w];
    pr[t] = acc;
    __syncthreads();
    if (g == 0)
        out_read[(size_t)b * W_ + w] =
            pr[w] + pr[64 + w] + pr[128 + w] + pr[192 + w];
}

extern "C" void kernel_launch(void* const* d_in, const int* in_sizes, int n_in,
                              void* d_out, int out_size, void* d_ws, size_t ws_size,
                              hipStream_t stream) {
    const float* r_key     = (const float*)d_in[0];
    const float* r_beta    = (const float*)d_in[1];
    const float* r_mode    = (const float*)d_in[2];
    const float* r_weights = (const float*)d_in[3];
    const float* memory    = (const float*)d_in[4];
    const float* link      = (const float*)d_in[5];
    float* out = (float*)d_out;                 // [read (B*W)] [weights (B*N)]

    float* cw = (float*)d_ws;                   // B*N
    float* fw = cw + (size_t)B_ * N_;           // B*N
    float* bw = fw + (size_t)B_ * N_;           // B*N

    content_kernel<<<B_, 512, 0, stream>>>(r_key, r_beta, memory, cw);
    link_matvec_kernel<<<B_ * 8, 256, 0, stream>>>(link, r_weights, fw, bw);
    combine_kernel<<<B_, 256, 0, stream>>>(r_mode, memory, cw, fw, bw,
                                           out, out + (size_t)B_ * W_);
}